// GPM_5274219839838
// MI455X (gfx1250) — compile-verified
//
#include <hip/hip_runtime.h>
#include <math.h>

// Problem constants (from reference)
#define EPS_    1e-6f
#define SIGMA2  0.25f          // SIGMA^2, SIGMA = 0.5
#define E_      64
#define B_      512
#define M_      128
#define C_      128
#define GSTR    132            // gram row stride (col M_ holds the RHS)
#define NT      256            // 8 wave32 waves

typedef __attribute__((ext_vector_type(2))) float v2f;
typedef __attribute__((ext_vector_type(8))) float v8f;

__global__ void gpm_zero_dkl(float* p) {
  if (threadIdx.x == 0) *p = 0.0f;
}

__global__ __launch_bounds__(NT) void gpm_write_kernel(
    const float* __restrict__ in_enc,     // (E,B,C) f32
    const float* __restrict__ mem_mean,   // (M,C)   f32
    const float* __restrict__ mem_logvar, // (1,)    f32
    float* __restrict__ out_mean,         // (B,M,C)
    float* __restrict__ out_cov,          // (B,M,M)
    float* __restrict__ out_dkl)          // scalar
{
  extern __shared__ float smem[];
  float* sm_mean = smem;                    // M*C        (64 KB)
  float* sm_gram = sm_mean + M_ * C_;       // M*GSTR     (66 KB), col M_ = rhs
  float* sm_L    = sm_gram + M_ * GSTR;     // E*M        (32 KB)  cov factor rows
  float* sm_z    = sm_L + E_ * M_;          // C
  float* sm_w    = sm_z + C_;               // M
  float* sm_d    = sm_w + M_;               // C  (delta)
  float* sm_c    = sm_d + C_;               // M  (u, then c = u/sig)
  float* sm_t    = sm_c + M_;               // E  (L w partials)
  float* sm_red  = sm_t + E_;               // NT (reduction)
  float* sm_s    = sm_red + NT;             // scalars

  const int tid  = threadIdx.x;
  const int b    = blockIdx.x;
  const int lane = tid & 31;
  const int wave = tid >> 5;        // 0..7 -> tile row
  const int ln   = lane & 15;       // within-tile column / row-M index
  const int lh   = lane >> 4;       // half-wave select

  const float pv = expf(mem_logvar[0]) + EPS_;   // prior variance (scalar)

  // prior mean -> LDS
  for (int idx = tid; idx < M_ * C_; idx += NT) sm_mean[idx] = mem_mean[idx];
  __syncthreads();

  for (int e = 0; e < E_; ++e) {
    // ---- load z_e, prefetch z_{e+1} -------------------------------------
    if (tid < C_) sm_z[tid] = in_enc[((size_t)e * B_ + b) * C_ + tid];
    if (e + 1 < E_ && tid == 0)
      __builtin_prefetch(&in_enc[((size_t)(e + 1) * B_ + b) * C_], 0, 3);
    __syncthreads();

    // ---- gram = mean * mean^T + SIGMA2*I  (V_WMMA_F32_16X16X4_F32) ------
    {
      const int ti = wave;
      for (int tj = 0; tj < 8; ++tj) {
        v8f acc = {0.f, 0.f, 0.f, 0.f, 0.f, 0.f, 0.f, 0.f};
        for (int kk = 0; kk < C_ / 4; ++kk) {
          const int k0 = kk * 4 + 2 * lh;
          v2f a, bv;
          a.x  = sm_mean[(ti * 16 + ln) * C_ + k0];
          a.y  = sm_mean[(ti * 16 + ln) * C_ + k0 + 1];
          bv.x = sm_mean[(tj * 16 + ln) * C_ + k0];
          bv.y = sm_mean[(tj * 16 + ln) * C_ + k0 + 1];
          acc = __builtin_amdgcn_wmma_f32_16x16x4_f32(
              false, a, false, bv, (short)0, acc, false, false);
        }
        for (int r = 0; r < 8; ++r) {
          const int gm = ti * 16 + r + 8 * lh;
          const int gn = tj * 16 + ln;
          float v = acc[r];
          if (gm == gn) v += SIGMA2;
          sm_gram[gm * GSTR + gn] = v;
        }
      }
    }
    // rhs: mz = mean * z  (column M_ of augmented gram)
    if (tid < M_) {
      float s = 0.f;
      for (int c = 0; c < C_; ++c) s += sm_mean[tid * C_ + c] * sm_z[c];
      sm_gram[tid * GSTR + M_] = s;
    }

    // ---- Gaussian elimination (SPD, no pivoting) on [gram | rhs] --------
    for (int k = 0; k < M_; ++k) {
      __syncthreads();
      const int i    = k + 1 + (tid >> 1);
      const int half = tid & 1;
      if (i < M_) {
        const float factor = sm_gram[i * GSTR + k] / sm_gram[k * GSTR + k];
        for (int j = k + 1 + half; j <= M_; j += 2)
          sm_gram[i * GSTR + j] -= factor * sm_gram[k * GSTR + j];
      }
    }
    // ---- back substitution (parallel column sweep) ----------------------
    for (int k = M_ - 1; k >= 0; --k) {
      __syncthreads();
      if (tid == 0)
        sm_w[k] = sm_gram[k * GSTR + M_] / sm_gram[k * GSTR + k];
      __syncthreads();
      if (tid < k)
        sm_gram[tid * GSTR + M_] -= sm_gram[tid * GSTR + k] * sm_w[k];
    }
    __syncthreads();

    // ---- delta = z - w^T mean ; t = L w (cov factor apply) --------------
    if (tid < C_) {
      float s = 0.f;
      for (int m = 0; m < M_; ++m) s += sm_w[m] * sm_mean[m * C_ + tid];
      sm_d[tid] = sm_z[tid] - s;
    } else {
      const int i = tid - C_;           // 0..127, need i < e (<64)
      if (i < e) {
        float s = 0.f;
        for (int m = 0; m < M_; ++m) s += sm_L[i * M_ + m] * sm_w[m];
        sm_t[i] = s;
      }
    }
    __syncthreads();
    // u = cov * w = pv*w - L^T (L w)
    if (tid < M_) {
      float u = pv * sm_w[tid];
      for (int i = 0; i < e; ++i) u -= sm_L[i * M_ + tid] * sm_t[i];
      sm_c[tid] = u;
    }
    __syncthreads();
    // sig = w^T u + SIGMA2
    if (tid == 0) {
      float s = SIGMA2;
      for (int m = 0; m < M_; ++m) s += sm_c[m] * sm_w[m];
      sm_s[0] = s;
    }
    __syncthreads();
    const float sig = sm_s[0];
    if (tid < M_) {
      const float u = sm_c[tid];
      sm_L[e * M_ + tid] = u / sqrtf(sig);   // cov downdate factor column
      sm_c[tid] = u / sig;                   // Kalman gain c
    }
    __syncthreads();
    // mean += c (outer) delta
    for (int idx = tid; idx < M_ * C_; idx += NT) {
      const int m = idx >> 7, c = idx & (C_ - 1);
      sm_mean[idx] += sm_c[m] * sm_d[c];
    }
    __syncthreads();
  }

  // ---- write post_mean --------------------------------------------------
  float* omean = out_mean + (size_t)b * M_ * C_;
  for (int idx = tid; idx < M_ * C_; idx += NT) omean[idx] = sm_mean[idx];

  // ---- post_cov = pv*I - L^T L  (V_WMMA_F32_16X16X4_F32, K = 64) --------
  float* ocov = out_cov + (size_t)b * M_ * M_;
  {
    const int ti = wave;
    for (int tj = 0; tj < 8; ++tj) {
      v8f acc = {0.f, 0.f, 0.f, 0.f, 0.f, 0.f, 0.f, 0.f};
      for (int kk = 0; kk < E_ / 4; ++kk) {
        const int k0 = kk * 4 + 2 * lh;
        v2f a, bv;
        a.x  = sm_L[(k0)     * M_ + ti * 16 + ln];
        a.y  = sm_L[(k0 + 1) * M_ + ti * 16 + ln];
        bv.x = sm_L[(k0)     * M_ + tj * 16 + ln];
        bv.y = sm_L[(k0 + 1) * M_ + tj * 16 + ln];
        acc = __builtin_amdgcn_wmma_f32_16x16x4_f32(
            false, a, false, bv, (short)0, acc, false, false);
      }
      for (int r = 0; r < 8; ++r) {
        const int gm = ti * 16 + r + 8 * lh;
        const int gn = tj * 16 + ln;
        float v = -acc[r];
        if (gm == gn) v += pv;
        ocov[gm * M_ + gn] = v;
      }
    }
  }

  // ---- dkl contribution -------------------------------------------------
  float part = 0.f;
  {
    const float inv_pv = 1.0f / pv;
    float t2p = 0.f;
    for (int idx = tid; idx < M_ * C_; idx += NT) {
      const float d = sm_mean[idx] - mem_mean[idx];
      t2p += d * d;
    }
    part = t2p * inv_pv;
    if (tid < M_) {
      float qd = pv;
      for (int i = 0; i < E_; ++i) { const float l = sm_L[i * M_ + tid]; qd -= l * l; }
      part += (float)C_ * (qd * inv_pv) + (float)C_ * (logf(pv) - logf(qd));
    }
  }
  sm_red[tid] = part;
  __syncthreads();
  for (int s = NT / 2; s > 0; s >>= 1) {
    if (tid < s) sm_red[tid] += sm_red[tid + s];
    __syncthreads();
  }
  if (tid == 0) {
    const float total = sm_red[0] - (float)C_ * (float)M_;   // + t3
    atomicAdd(out_dkl, total * (1.0f / (float)B_));          // mean over batch
  }
}

extern "C" void kernel_launch(void* const* d_in, const int* in_sizes, int n_in,
                              void* d_out, int out_size, void* d_ws, size_t ws_size,
                              hipStream_t stream) {
  (void)in_sizes; (void)n_in; (void)out_size; (void)d_ws; (void)ws_size;

  const float* in_enc     = (const float*)d_in[0];   // (E,B,C)
  const float* mem_mean   = (const float*)d_in[1];   // (M,C)
  const float* mem_logvar = (const float*)d_in[2];   // (1,)

  float* out_mean = (float*)d_out;
  float* out_cov  = out_mean + (size_t)B_ * M_ * C_;
  float* out_dkl  = out_cov  + (size_t)B_ * M_ * M_;

  const size_t smem_floats =
      (size_t)M_ * C_ + (size_t)M_ * GSTR + (size_t)E_ * M_ +
      C_ + M_ + C_ + M_ + E_ + NT + 8;
  const size_t smem = smem_floats * sizeof(float);   // ~165 KB < 320 KB/WGP

  // Unconditional, idempotent, deterministic — no static guards (harness
  // contract). Not a stream op, so it is safe under graph capture.
  hipFuncSetAttribute(reinterpret_cast<const void*>(gpm_write_kernel),
                      hipFuncAttributeMaxDynamicSharedMemorySize, (int)smem);

  gpm_zero_dkl<<<1, 32, 0, stream>>>(out_dkl);
  gpm_write_kernel<<<dim3(B_), dim3(NT), smem, stream>>>(
      in_enc, mem_mean, mem_logvar, out_mean, out_cov, out_dkl);
}